// Inferencer_9423158248217
// MI455X (gfx1250) — compile-verified
//
#include <hip/hip_runtime.h>
#include <hip/hip_bf16.h>

// ---------------- problem constants (match reference) ----------------
#define N_NODES   100000
#define N_EDGES   1600000
#define NFEAT     256
#define NHID      64
#define NHEADS    8
#define NHIDT     (NHEADS * NHID)   // 512
#define NCLASS    97
#define NCLASS_P  112               // padded to 7*16 for WMMA tiles
#define ALPHA     0.2f
#define EPS_GAT   1e-16f

typedef __attribute__((ext_vector_type(2))) float v2f;
typedef __attribute__((ext_vector_type(8))) float v8f;

// ---------------- zero fill ----------------
__global__ __launch_bounds__(256) void k_zero(float* __restrict__ p, size_t n) {
    size_t i = (size_t)blockIdx.x * blockDim.x + threadIdx.x;
    size_t stride = (size_t)gridDim.x * blockDim.x;
    for (; i < n; i += stride) p[i] = 0.0f;
}

// ---------------- GEMM1: H[N,512] = X[N,256] @ Wall, fp32 WMMA 16x16x4 ----------------
// W is [8][256][64]; output column c belongs to head c>>6, inner col c&63.
// One wave per 16x16 output tile; K loop in steps of 4 -> v_wmma_f32_16x16x4_f32.
__global__ __launch_bounds__(256) void k_gemm_hidden(const float* __restrict__ X,
                                                     const float* __restrict__ W,
                                                     float* __restrict__ H) {
    const int tilesN = NHIDT / 16;                         // 32
    const int totalTiles = (N_NODES / 16) * tilesN;        // 6250*32
    int tile = blockIdx.x * (blockDim.x >> 5) + (threadIdx.x >> 5);
    if (tile >= totalTiles) return;                        // wave-uniform exit
    int tm = tile / tilesN, tn = tile % tilesN;
    int lane = threadIdx.x & 31;
    int half = lane >> 4;        // 0: K pair {0,1}; 1: K pair {2,3}
    int l    = lane & 15;

    int rowA = tm * 16 + l;
    int col  = tn * 16 + l;
    int head = (tn * 16) >> 6;                             // whole tile in one head
    int cc   = col & 63;
    const float* __restrict__ ap = X + (size_t)rowA * NFEAT + half * 2;
    const float* __restrict__ wb = W + (size_t)head * (NFEAT * NHID) + cc + (size_t)(half * 2) * NHID;

    v8f acc = {};
    for (int k = 0; k < NFEAT; k += 4) {
        if ((k & 63) == 0) __builtin_prefetch(ap + k + 128, 0, 1);   // global_prefetch_b8
        v2f a = *reinterpret_cast<const v2f*>(ap + k);               // A 16x4 f32 lane layout
        v2f b;
        b.x = wb[(size_t)k * NHID];                                  // B rows k+2h, k+2h+1
        b.y = wb[(size_t)(k + 1) * NHID];
        acc = __builtin_amdgcn_wmma_f32_16x16x4_f32(false, a, false, b,
                                                    (short)0, acc, false, false);
    }
    // D layout: VGPR j -> row tm*16 + j + half*8, col tn*16 + (lane&15)
    #pragma unroll
    for (int j = 0; j < 8; ++j) {
        int row = tm * 16 + j + half * 8;
        H[(size_t)row * NHIDT + col] = acc[j];
    }
}

// ---------------- layer-1 attention scores ----------------
__global__ __launch_bounds__(256) void k_scores1(const float* __restrict__ H,
                                                 const float* __restrict__ a,
                                                 float* __restrict__ ssrc,
                                                 float* __restrict__ sdst) {
    int t = blockIdx.x * blockDim.x + threadIdx.x;
    if (t >= N_NODES * NHEADS) return;
    int n = t >> 3, k = t & 7;
    const float* h  = H + (size_t)n * NHIDT + k * NHID;
    const float* a1 = a + k * (2 * NHID);
    const float* a2 = a1 + NHID;
    float s1 = 0.f, s2 = 0.f;
    #pragma unroll 8
    for (int j = 0; j < NHID; ++j) { float hv = h[j]; s1 += hv * a1[j]; s2 += hv * a2[j]; }
    ssrc[t] = s1; sdst[t] = s2;
}

// ---------------- layer-1 edge scatter: one wave per (edge, head) ----------------
__global__ __launch_bounds__(256) void k_edge1(const int* __restrict__ ei,
                                               const float* __restrict__ H,
                                               const float* __restrict__ ssrc,
                                               const float* __restrict__ sdst,
                                               float* __restrict__ num,
                                               float* __restrict__ rowsum) {
    long long wid = ((long long)blockIdx.x * blockDim.x + threadIdx.x) >> 5;
    if (wid >= (long long)N_EDGES * NHEADS) return;        // wave-uniform
    int lane = threadIdx.x & 31;
    int e = (int)(wid >> 3), k = (int)(wid & 7);
    int s = ei[e], d = ei[N_EDGES + e];
    float sc = ssrc[s * NHEADS + k] + sdst[d * NHEADS + k];
    float lr = sc > 0.f ? sc : ALPHA * sc;
    float w  = expf(-lr);
    const float* hd = H   + (size_t)d * NHIDT + k * NHID;
    float*       np = num + (size_t)s * NHIDT + k * NHID;
    atomicAdd(np + lane,      w * hd[lane]);
    atomicAdd(np + lane + 32, w * hd[lane + 32]);
    if (lane == 0) atomicAdd(rowsum + (size_t)s * NHEADS + k, w);
}

// ---------------- layer-1 finalize: x = elu(num / (rowsum+eps)), in place ----------------
__global__ __launch_bounds__(256) void k_final1(float* __restrict__ num,
                                                const float* __restrict__ rowsum) {
    size_t t = (size_t)blockIdx.x * blockDim.x + threadIdx.x;
    if (t >= (size_t)N_NODES * NHIDT) return;
    size_t n = t >> 9;            // /512
    int    k = (int)((t >> 6) & 7);
    float rs = rowsum[n * NHEADS + k] + EPS_GAT;
    float v  = num[t] / rs;
    num[t] = v > 0.f ? v : (expf(v) - 1.f);
}

// ---------------- GEMM2: H2[N,112] = Xcat[N,512] @ W_out[512,97] (zero-padded cols) ----------------
__global__ __launch_bounds__(256) void k_gemm_out(const float* __restrict__ Xc,
                                                  const float* __restrict__ Wo,
                                                  float* __restrict__ H2) {
    const int tilesN = NCLASS_P / 16;                      // 7
    const int totalTiles = (N_NODES / 16) * tilesN;
    int tile = blockIdx.x * (blockDim.x >> 5) + (threadIdx.x >> 5);
    if (tile >= totalTiles) return;
    int tm = tile / tilesN, tn = tile % tilesN;
    int lane = threadIdx.x & 31;
    int half = lane >> 4;
    int l    = lane & 15;

    int rowA = tm * 16 + l;
    int col  = tn * 16 + l;
    bool cvalid = (col < NCLASS);
    const float* __restrict__ ap = Xc + (size_t)rowA * NHIDT + half * 2;
    const float* __restrict__ wb = Wo + (size_t)(half * 2) * NCLASS + (cvalid ? col : 0);

    v8f acc = {};
    for (int k = 0; k < NHIDT; k += 4) {
        if ((k & 63) == 0) __builtin_prefetch(ap + k + 128, 0, 1);
        v2f a = *reinterpret_cast<const v2f*>(ap + k);
        v2f b;
        b.x = cvalid ? wb[(size_t)k * NCLASS]       : 0.f;
        b.y = cvalid ? wb[(size_t)(k + 1) * NCLASS] : 0.f;
        acc = __builtin_amdgcn_wmma_f32_16x16x4_f32(false, a, false, b,
                                                    (short)0, acc, false, false);
    }
    #pragma unroll
    for (int j = 0; j < 8; ++j) {
        int row = tm * 16 + j + half * 8;
        H2[(size_t)row * NCLASS_P + col] = acc[j];         // cols>=97 are exact zeros
    }
}

// ---------------- layer-2 attention scores ----------------
__global__ __launch_bounds__(256) void k_scores2(const float* __restrict__ H2,
                                                 const float* __restrict__ aout,
                                                 float* __restrict__ ssrc,
                                                 float* __restrict__ sdst) {
    int n = blockIdx.x * blockDim.x + threadIdx.x;
    if (n >= N_NODES) return;
    const float* h = H2 + (size_t)n * NCLASS_P;
    float s1 = 0.f, s2 = 0.f;
    for (int j = 0; j < NCLASS; ++j) { float hv = h[j]; s1 += hv * aout[j]; s2 += hv * aout[NCLASS + j]; }
    ssrc[n] = s1; sdst[n] = s2;
}

// ---------------- layer-2 edge scatter: one wave per edge, accumulate into d_out ----------------
__global__ __launch_bounds__(256) void k_edge2(const int* __restrict__ ei,
                                               const float* __restrict__ H2,
                                               const float* __restrict__ ssrc,
                                               const float* __restrict__ sdst,
                                               float* __restrict__ out,
                                               float* __restrict__ rowsum2) {
    long long wid = ((long long)blockIdx.x * blockDim.x + threadIdx.x) >> 5;
    if (wid >= (long long)N_EDGES) return;
    int lane = threadIdx.x & 31;
    int e = (int)wid;
    int s = ei[e], d = ei[N_EDGES + e];
    float sc = ssrc[s] + sdst[d];
    float lr = sc > 0.f ? sc : ALPHA * sc;
    float w  = expf(-lr);
    const float* hd = H2  + (size_t)d * NCLASS_P;
    float*       op = out + (size_t)s * NCLASS;
    for (int j = lane; j < NCLASS; j += 32)
        atomicAdd(op + j, w * hd[j]);
    if (lane == 0) atomicAdd(rowsum2 + s, w);
}

// ---------------- layer-2 finalize: elu + log_softmax, one wave per node, in place ----------------
__global__ __launch_bounds__(256) void k_final2(float* __restrict__ out,
                                                const float* __restrict__ rowsum2) {
    int wid = (blockIdx.x * blockDim.x + threadIdx.x) >> 5;
    if (wid >= N_NODES) return;
    int lane = threadIdx.x & 31;
    float rs = rowsum2[wid] + EPS_GAT;
    float* row = out + (size_t)wid * NCLASS;

    float v[4];
    float m = -3.402823466e38f;
    #pragma unroll
    for (int i = 0; i < 4; ++i) {
        int j = lane + 32 * i;
        if (j < NCLASS) {
            float x = row[j] / rs;
            v[i] = x > 0.f ? x : (expf(x) - 1.f);          // elu
            m = fmaxf(m, v[i]);
        }
    }
    #pragma unroll
    for (int off = 16; off > 0; off >>= 1) m = fmaxf(m, __shfl_xor(m, off, 32));

    float se = 0.f;
    #pragma unroll
    for (int i = 0; i < 4; ++i) {
        int j = lane + 32 * i;
        if (j < NCLASS) se += expf(v[i] - m);
    }
    #pragma unroll
    for (int off = 16; off > 0; off >>= 1) se += __shfl_xor(se, off, 32);
    float lse = logf(se);

    #pragma unroll
    for (int i = 0; i < 4; ++i) {
        int j = lane + 32 * i;
        if (j < NCLASS) row[j] = v[i] - m - lse;
    }
}

// ---------------- launcher ----------------
extern "C" void kernel_launch(void* const* d_in, const int* in_sizes, int n_in,
                              void* d_out, int out_size, void* d_ws, size_t ws_size,
                              hipStream_t stream) {
    const float* features = (const float*)d_in[0];   // [N, 256]
    const float* W        = (const float*)d_in[1];   // [8, 256, 64]
    const float* a        = (const float*)d_in[2];   // [8, 128]
    const float* W_out    = (const float*)d_in[3];   // [512, 97]
    const float* a_out    = (const float*)d_in[4];   // [194]
    const int*   ei       = (const int*)d_in[5];     // [2, E]
    float* out = (float*)d_out;                      // [N, 97]
    (void)in_sizes; (void)n_in; (void)out_size; (void)ws_size;

    // workspace layout (fp32 elements)
    float* ws = (float*)d_ws;
    size_t oH   = 0;
    size_t oS1  = oH   + (size_t)N_NODES * NHIDT;    // s_src  [N,8]
    size_t oD1  = oS1  + (size_t)N_NODES * NHEADS;   // s_dst  [N,8]
    size_t oRS1 = oD1  + (size_t)N_NODES * NHEADS;   // rowsum [N,8]
    size_t oNUM = oRS1 + (size_t)N_NODES * NHEADS;   // num/x  [N,512]
    size_t oH2  = oNUM + (size_t)N_NODES * NHIDT;    // H2     [N,112]
    size_t oS2  = oH2  + (size_t)N_NODES * NCLASS_P; // s2_src [N]
    size_t oD2  = oS2  + (size_t)N_NODES;            // s2_dst [N]
    size_t oRS2 = oD2  + (size_t)N_NODES;            // rowsum2[N]
    float* H    = ws + oH;
    float* s1   = ws + oS1;
    float* d1   = ws + oD1;
    float* rs1  = ws + oRS1;
    float* num  = ws + oNUM;
    float* H2   = ws + oH2;
    float* s2   = ws + oS2;
    float* d2   = ws + oD2;
    float* rs2  = ws + oRS2;

    // zero accumulators (harness does not re-poison between replays)
    k_zero<<<2048, 256, 0, stream>>>(rs1, (size_t)N_NODES * NHEADS);
    k_zero<<<4096, 256, 0, stream>>>(num, (size_t)N_NODES * NHIDT);
    k_zero<<<1024, 256, 0, stream>>>(rs2, (size_t)N_NODES);
    k_zero<<<4096, 256, 0, stream>>>(out, (size_t)N_NODES * NCLASS);

    // layer 1
    {
        int tiles = (N_NODES / 16) * (NHIDT / 16);
        k_gemm_hidden<<<(tiles + 7) / 8, 256, 0, stream>>>(features, W, H);
    }
    k_scores1<<<(N_NODES * NHEADS + 255) / 256, 256, 0, stream>>>(H, a, s1, d1);
    {
        long long waves = (long long)N_EDGES * NHEADS;
        long long blocks = (waves * 32 + 255) / 256;
        k_edge1<<<(int)blocks, 256, 0, stream>>>(ei, H, s1, d1, num, rs1);
    }
    {
        size_t t = (size_t)N_NODES * NHIDT;
        k_final1<<<(int)((t + 255) / 256), 256, 0, stream>>>(num, rs1);
    }

    // layer 2
    {
        int tiles = (N_NODES / 16) * (NCLASS_P / 16);
        k_gemm_out<<<(tiles + 7) / 8, 256, 0, stream>>>(num, W_out, H2);
    }
    k_scores2<<<(N_NODES + 255) / 256, 256, 0, stream>>>(H2, a_out, s2, d2);
    {
        long long blocks = ((long long)N_EDGES * 32 + 255) / 256;
        k_edge2<<<(int)blocks, 256, 0, stream>>>(ei, H2, s2, d2, out, rs2);
    }
    k_final2<<<(N_NODES * 32 + 255) / 256, 256, 0, stream>>>(out, rs2);
}